// TERMNeighborAttention_20847771255441
// MI455X (gfx1250) — compile-verified
//
#include <hip/hip_runtime.h>

typedef float v2f __attribute__((ext_vector_type(2)));
typedef float v8f __attribute__((ext_vector_type(8)));

#define NUM_HIDDEN 128
#define NUM_IN     256
#define NUM_HEADS  4
#define HEAD_DIM   32
#define KNBR       30
#define NTOK       12288   // B*T*N = 4*32*96
#define TOKPB      4       // tokens per workgroup in attn_fused (16 wave32)
#define ATTN_THREADS (TOKPB * NUM_HEADS * 32)   // 512

// -----------------------------------------------------------------------------
// Generic fp32 WMMA GEMM:  C[M x 128] = A[M x 128] @ B[128 x 128]
// 256 threads (8 wave32), 64 rows of C per workgroup.
// -----------------------------------------------------------------------------
__global__ __launch_bounds__(256) void gemm_f32_wmma(const float* __restrict__ A,
                                                     const float* __restrict__ B,
                                                     float* __restrict__ C, int M)
{
    __shared__ float sA[64][36];    // 64 rows x 32-col K-chunk (+pad)
    __shared__ float sB[32][132];   // K-chunk x 128 cols (+pad)

    const int tid  = threadIdx.x;
    const int wave = tid >> 5;
    const int lane = tid & 31;
    const int half = lane >> 4;
    const int l16  = lane & 15;

    const int mtile   = wave & 3;
    const int ngroup  = wave >> 2;
    const int rowBase = blockIdx.x * 64;

    v8f acc[4] = {};

    for (int kc = 0; kc < NUM_HIDDEN; kc += 32) {
        for (int idx = tid; idx < 64 * 8; idx += 256) {
            const int r = idx >> 3, c4 = (idx & 7) << 2;
            const float4 v = *(const float4*)&A[(size_t)(rowBase + r) * NUM_HIDDEN + kc + c4];
            sA[r][c4 + 0] = v.x; sA[r][c4 + 1] = v.y;
            sA[r][c4 + 2] = v.z; sA[r][c4 + 3] = v.w;
        }
        for (int idx = tid; idx < 32 * 32; idx += 256) {
            const int r = idx >> 5, c4 = (idx & 31) << 2;
            const float4 v = *(const float4*)&B[(size_t)(kc + r) * NUM_HIDDEN + c4];
            sB[r][c4 + 0] = v.x; sB[r][c4 + 1] = v.y;
            sB[r][c4 + 2] = v.z; sB[r][c4 + 3] = v.w;
        }
        __syncthreads();

        for (int k = 0; k < 32; k += 4) {
            v2f a;
            const int am = mtile * 16 + l16;
            a.x = sA[am][k + 2 * half + 0];
            a.y = sA[am][k + 2 * half + 1];
            #pragma unroll
            for (int t = 0; t < 4; ++t) {
                const int n = (ngroup * 4 + t) * 16 + l16;
                v2f b;
                b.x = sB[k + 2 * half + 0][n];
                b.y = sB[k + 2 * half + 1][n];
                acc[t] = __builtin_amdgcn_wmma_f32_16x16x4_f32(
                             false, a, false, b, (short)0, acc[t], false, false);
            }
        }
        __syncthreads();
    }

    #pragma unroll
    for (int t = 0; t < 4; ++t) {
        const int n = (ngroup * 4 + t) * 16 + l16;
        #pragma unroll
        for (int i = 0; i < 8; ++i) {
            const int m = rowBase + mtile * 16 + half * 8 + i;
            C[(size_t)m * NUM_HIDDEN + n] = acc[t][i];
        }
    }
}

// -----------------------------------------------------------------------------
// Fused neighbor attention, 4 tokens per workgroup (512 threads = 16 wave32),
// K/V register-resident. wave w -> (token tk = w>>2, head h = w&3). Each wave
// computes via WMMA the 2x2 Kp tiles (cols h*32..+31 of E@Wk) and 2x2 V tiles
// (same cols of E@Wv) for its token's 32 padded neighbor rows, keeps them in
// 64 accumulator VGPRs, then does masked softmax + attend.V with shuffles.
// E is staged in LDS in 64-wide K chunks (conflict-free padded stride).
// -----------------------------------------------------------------------------
__global__ __launch_bounds__(ATTN_THREADS) void attn_fused(
        const float* __restrict__ hEV,
        const int*   __restrict__ maskAtt,
        const float* __restrict__ Q,
        const float* __restrict__ Wk,
        const float* __restrict__ Wv,
        float* __restrict__ Ctx)
{
    __shared__ float sE[TOKPB * 32][68];     // 4 tokens x 32 padded rows, 64-col K chunk
    __shared__ float sQ[TOKPB][NUM_HIDDEN];
    __shared__ float sLog[TOKPB * NUM_HEADS][32];
    __shared__ float sAtt[TOKPB * NUM_HEADS][32];
    __shared__ int   sMask[TOKPB][32];

    const int tok0 = blockIdx.x * TOKPB;
    const int tid  = threadIdx.x;
    const int wave = tid >> 5;
    const int lane = tid & 31;
    const int half = lane >> 4;
    const int l16  = lane & 15;

    const int h  = wave & 3;              // head
    const int tk = wave >> 2;             // token within block

    // ---- Q rows + mask ------------------------------------------------------
    {
        const int qt = tid >> 7, qc = tid & 127;           // one float per thread
        sQ[qt][qc] = Q[(size_t)(tok0 + qt) * NUM_HIDDEN + qc];
        if (tid < TOKPB * 32) {
            const int mt = tid >> 5, mk = tid & 31;
            sMask[mt][mk] = (mk < KNBR) ? maskAtt[(size_t)(tok0 + mt) * KNBR + mk] : 0;
        }
    }

    v8f accK[2][2] = {};                  // [mtile][Kp n-tile]
    v8f accV[2][2] = {};                  // [mtile][V  n-tile]

    // ---- KV projection: stream E in 64-wide K chunks ------------------------
    for (int kc = 0; kc < NUM_IN; kc += 64) {
        __syncthreads();                  // previous chunk fully consumed
        for (int idx = tid; idx < TOKPB * 32 * 16; idx += ATTN_THREADS) { // float4 granules
            const int r = idx >> 4, c4 = (idx & 15) << 2;
            const int tk2 = r >> 5, lr = r & 31;
            float4 v = make_float4(0.f, 0.f, 0.f, 0.f);
            if (lr < KNBR)
                v = *(const float4*)&hEV[((size_t)(tok0 + tk2) * KNBR + lr) * NUM_IN + kc + c4];
            sE[r][c4 + 0] = v.x; sE[r][c4 + 1] = v.y;
            sE[r][c4 + 2] = v.z; sE[r][c4 + 3] = v.w;
        }
        __syncthreads();

        for (int k = 0; k < 64; k += 4) {
            // A fragments for this wave's token (rows tk*32 .. +31)
            v2f a0, a1;
            const int r0 = tk * 32 + l16;
            a0.x = sE[r0][k + 2 * half + 0];
            a0.y = sE[r0][k + 2 * half + 1];
            a1.x = sE[r0 + 16][k + 2 * half + 0];
            a1.y = sE[r0 + 16][k + 2 * half + 1];

            const int kg = kc + k + 2 * half;              // global K row
            #pragma unroll
            for (int t = 0; t < 2; ++t) {
                const int col = h * HEAD_DIM + t * 16 + l16;
                v2f bk, bv;
                bk.x = Wk[(size_t)(kg + 0) * NUM_HIDDEN + col];
                bk.y = Wk[(size_t)(kg + 1) * NUM_HIDDEN + col];
                bv.x = Wv[(size_t)(kg + 0) * NUM_HIDDEN + col];
                bv.y = Wv[(size_t)(kg + 1) * NUM_HIDDEN + col];
                accK[0][t] = __builtin_amdgcn_wmma_f32_16x16x4_f32(
                                 false, a0, false, bk, (short)0, accK[0][t], false, false);
                accK[1][t] = __builtin_amdgcn_wmma_f32_16x16x4_f32(
                                 false, a1, false, bk, (short)0, accK[1][t], false, false);
                accV[0][t] = __builtin_amdgcn_wmma_f32_16x16x4_f32(
                                 false, a0, false, bv, (short)0, accV[0][t], false, false);
                accV[1][t] = __builtin_amdgcn_wmma_f32_16x16x4_f32(
                                 false, a1, false, bv, (short)0, accV[1][t], false, false);
            }
        }
    }

    // ---- logits: logit[k] = sum_n Q[h,n] * Kp[k,n] (C-layout aware) ---------
    // C element (m,n) lives in VGPR m%8, lane (m/8)*16 + n.
    const float q0 = sQ[tk][h * HEAD_DIM + l16];
    const float q1 = sQ[tk][h * HEAD_DIM + 16 + l16];
    #pragma unroll
    for (int m = 0; m < 2; ++m) {
        #pragma unroll
        for (int i = 0; i < 8; ++i) {
            float p = accK[m][0][i] * q0 + accK[m][1][i] * q1;
            #pragma unroll
            for (int off = 8; off >= 1; off >>= 1)
                p += __shfl_xor(p, off, 32);               // reduce within 16-lane half
            if (l16 == 0)                                   // lanes 0 and 16 write
                sLog[wave][m * 16 + half * 8 + i] = p;
        }
    }
    __syncthreads();                                        // uniform across WG

    // ---- masked softmax (lane = neighbor index) -----------------------------
    const bool validk = (lane < KNBR) && (sMask[tk][lane] > 0);
    float logit = validk ? sLog[wave][lane] * 0.17677669529663687f   // 1/sqrt(32)
                         : -3.402823466e38f;                          // finfo(f32).min
    float mx = logit;
    #pragma unroll
    for (int off = 16; off >= 1; off >>= 1)
        mx = fmaxf(mx, __shfl_xor(mx, off, 32));
    const float e = __expf(logit - mx);
    float s = e;
    #pragma unroll
    for (int off = 16; off >= 1; off >>= 1)
        s += __shfl_xor(s, off, 32);
    sAtt[wave][lane] = validk ? (e / s) : 0.f;              // attend * mask
    __syncthreads();                                        // uniform across WG

    // ---- context: out[j] = sum_k att[k] * V[k][j], V still in registers -----
    float pj[2];
    #pragma unroll
    for (int t = 0; t < 2; ++t) {
        float p = 0.f;
        #pragma unroll
        for (int m = 0; m < 2; ++m) {
            #pragma unroll
            for (int i = 0; i < 8; ++i)
                p += sAtt[wave][m * 16 + half * 8 + i] * accV[m][t][i];
        }
        p += __shfl_xor(p, 16, 32);                         // combine lane halves
        pj[t] = p;
    }
    const float outv = half ? pj[1] : pj[0];
    Ctx[(size_t)(tok0 + tk) * NUM_HIDDEN + h * HEAD_DIM + half * 16 + l16] = outv;
}

// -----------------------------------------------------------------------------
extern "C" void kernel_launch(void* const* d_in, const int* in_sizes, int n_in,
                              void* d_out, int out_size, void* d_ws, size_t ws_size,
                              hipStream_t stream)
{
    const float* h_V  = (const float*)d_in[0];
    const float* h_EV = (const float*)d_in[1];
    const int*   mask = (const int*)d_in[2];
    const float* Wq   = (const float*)d_in[3];
    const float* Wk   = (const float*)d_in[4];
    const float* Wv   = (const float*)d_in[5];
    const float* Wo   = (const float*)d_in[6];

    float* gQ   = (float*)d_ws;                          // 12288*128 f32
    float* gCtx = gQ + (size_t)NTOK * NUM_HIDDEN;        // 12288*128 f32

    // 1) Q = h_V @ Wq
    gemm_f32_wmma<<<NTOK / 64, 256, 0, stream>>>(h_V, Wq, gQ, NTOK);
    // 2) fused K/V projection + masked softmax + attend.V (streams h_EV once)
    attn_fused<<<NTOK / TOKPB, ATTN_THREADS, 0, stream>>>(h_EV, mask, gQ, Wk, Wv, gCtx);
    // 3) out = ctx @ Wo
    gemm_f32_wmma<<<NTOK / 64, 256, 0, stream>>>(gCtx, Wo, (float*)d_out, NTOK);
}